// FASINet_74371653698080
// MI455X (gfx1250) — compile-verified
//
#include <hip/hip_runtime.h>
#include <hip/hip_bf16.h>
#include <math.h>

// Problem constants
#define BATCH 4
#define DIM   64
#define HH    256
#define WW    256
#define HEADS 8
#define CPH   8                 // DIM / HEADS
#define DFEAT 2048              // HH * CPH
#define HW    (HH * WW)         // 65536
#define PLANE ((long)DIM * HW)  // elems per batch image = 4,194,304
#define NTOT  ((long)BATCH * PLANE) // 16,777,216 per tensor
#define KC    32                // K-chunk staged in LDS for the big GEMMs

typedef float v2f __attribute__((ext_vector_type(2)));
typedef float v8f __attribute__((ext_vector_type(8)));
typedef int   v4i __attribute__((ext_vector_type(4)));

typedef __attribute__((address_space(1))) v4i gv4i;   // global int4
typedef __attribute__((address_space(3))) v4i lv4i;   // LDS int4

// D = A(16x4 f32) * B(4x16 f32) + C  -- full-precision WMMA on gfx1250
static __device__ __forceinline__ v8f wmma4(v2f a, v2f b, v8f c) {
  return __builtin_amdgcn_wmma_f32_16x16x4_f32(false, a, false, b, (short)0, c,
                                               false, false);
}

// Async global -> LDS copy of 16 bytes per lane (ASYNCcnt-tracked).
static __device__ __forceinline__ void async_ld16(const float* g, float* l) {
#if __has_builtin(__builtin_amdgcn_global_load_async_to_lds_b128)
  __builtin_amdgcn_global_load_async_to_lds_b128((gv4i*)g, (lv4i*)l, 0, 0);
#else
  unsigned lo = (unsigned)(unsigned long long)l;           // LDS byte offset
  asm volatile("global_load_async_to_lds_b128 %0, %1, off"
               :: "v"(lo), "v"((unsigned long long)g) : "memory");
#endif
}

static __device__ __forceinline__ void wait_async0() {
#if __has_builtin(__builtin_amdgcn_s_wait_asynccnt)
  __builtin_amdgcn_s_wait_asynccnt(0);
#else
  asm volatile("s_wait_asynccnt 0x0" ::: "memory");
#endif
}

// ---------------------------------------------------------------------------
// 1) LayerNorm over channels, per pixel, both tensors at once.
// ---------------------------------------------------------------------------
__global__ void ln_kernel(const float* __restrict__ x1, const float* __restrict__ x2,
                          const float* __restrict__ ln1w, const float* __restrict__ ln1b,
                          const float* __restrict__ ln2w, const float* __restrict__ ln2b,
                          float* __restrict__ x1n, float* __restrict__ x2n) {
  int p   = blockIdx.x * blockDim.x + threadIdx.x;   // [0, BATCH*HW)
  int b   = p >> 16;                                 // / HW
  int rem = p & (HW - 1);
  long base = (long)b * PLANE + rem;
  float s1 = 0.f, q1 = 0.f, s2 = 0.f, q2 = 0.f;
  for (int c = 0; c < DIM; ++c) {
    float a = x1[base + (long)c * HW]; s1 += a; q1 += a * a;
    float e = x2[base + (long)c * HW]; s2 += e; q2 += e * e;
  }
  const float inv = 1.0f / DIM;
  float mu1 = s1 * inv, var1 = q1 * inv - mu1 * mu1;
  float mu2 = s2 * inv, var2 = q2 * inv - mu2 * mu2;
  float r1 = rsqrtf(var1 + 1e-5f), r2 = rsqrtf(var2 + 1e-5f);
  for (int c = 0; c < DIM; ++c) {
    long i = base + (long)c * HW;
    x1n[i] = (x1[i] - mu1) * r1 * ln1w[c] + ln1b[c];
    x2n[i] = (x2[i] - mu2) * r2 * ln2w[c] + ln2b[c];
  }
}

// ---------------------------------------------------------------------------
// 2) Sum of 6 depthwise convs == one 21-tap horizontal + one 21-tap vertical
//    pass with per-channel combined weights (zero padding).
// ---------------------------------------------------------------------------
__global__ void dw_kernel(const float* __restrict__ xin, float* __restrict__ dout,
                          const float* __restrict__ kh7,  const float* __restrict__ kh11,
                          const float* __restrict__ kh21, const float* __restrict__ kv7,
                          const float* __restrict__ kv11, const float* __restrict__ kv21,
                          const float* __restrict__ bdw /* 6 rows of DIM */) {
  __shared__ float wh[21], wv[21], sb[1];
  int c = blockIdx.y & (DIM - 1);
  if (threadIdx.x < 21) {
    int t = threadIdx.x;
    float h = kh21[c * 21 + t];
    if (t >= 5 && t < 16) h += kh11[c * 11 + (t - 5)];
    if (t >= 7 && t < 14) h += kh7 [c * 7  + (t - 7)];
    wh[t] = h;
    float v = kv21[c * 21 + t];
    if (t >= 5 && t < 16) v += kv11[c * 11 + (t - 5)];
    if (t >= 7 && t < 14) v += kv7 [c * 7  + (t - 7)];
    wv[t] = v;
  }
  if (threadIdx.x == 0) {
    float s = 0.f;
    for (int i = 0; i < 6; ++i) s += bdw[i * DIM + c];
    sb[0] = s;
  }
  __syncthreads();

  int h = blockIdx.x, w = threadIdx.x;
  const float* plane = xin + (long)blockIdx.y * HW;
  const float* row   = plane + (long)h * WW;
  float acc = sb[0];
  for (int j = 0; j < 21; ++j) {
    int wp = w + j - 10;
    if ((unsigned)wp < WW) acc += wh[j] * row[wp];
  }
  for (int j = 0; j < 21; ++j) {
    int hp = h + j - 10;
    if ((unsigned)hp < HH) acc += wv[j] * plane[(long)hp * WW + w];
  }
  dout[(long)blockIdx.y * HW + (long)h * WW + w] = acc;
}

// ---------------------------------------------------------------------------
// 3) Pointwise 64x64 conv as WMMA GEMM; output written in attention layout
//    (b, head, w, h*cph).  Block = 8 waves: 4 m-tiles x 2 n-halves, 32 pixels.
// ---------------------------------------------------------------------------
__global__ void pw_attn_kernel(const float* __restrict__ din,
                               const float* __restrict__ wpw,
                               const float* __restrict__ bpw,
                               float* __restrict__ pout) {
  int lane = threadIdx.x & 31, wvid = threadIdx.x >> 5;
  int mtile = wvid & 3, nhalf = wvid >> 2;
  int w0 = blockIdx.x * 32 + nhalf * 16;
  int h  = blockIdx.y;
  int b  = blockIdx.z;
  int m0 = mtile * 16;
  int lm = lane & 15, kb = (lane >> 4) << 1;
  const float* src = din + (long)b * PLANE + (long)h * WW + w0;
  v8f acc = {};
  for (int k0 = 0; k0 < DIM; k0 += 4) {
    v2f a, bb;
    a.x  = wpw[(m0 + lm) * DIM + k0 + kb];
    a.y  = wpw[(m0 + lm) * DIM + k0 + kb + 1];
    bb.x = src[(long)(k0 + kb)     * HW + lm];
    bb.y = src[(long)(k0 + kb + 1) * HW + lm];
    acc = wmma4(a, bb, acc);
  }
  int wn = w0 + lm;
  for (int r = 0; r < 8; ++r) {
    int o = m0 + r + ((lane >> 4) << 3);
    long idx = ((long)(b * HEADS + (o >> 3)) * WW + wn) * DFEAT + h * CPH + (o & 7);
    pout[idx] = acc[r] + bpw[o];
  }
}

// ---------------------------------------------------------------------------
// 4) Per-row inverse L2 norm (rows of length DFEAT).
// ---------------------------------------------------------------------------
__global__ void rownorm_kernel(const float* __restrict__ p, float* __restrict__ invn) {
  __shared__ float red[256];
  const float* r = p + (long)blockIdx.x * DFEAT;
  float s = 0.f;
  for (int i = threadIdx.x; i < DFEAT; i += 256) { float v = r[i]; s += v * v; }
  red[threadIdx.x] = s; __syncthreads();
  for (int st = 128; st; st >>= 1) {
    if (threadIdx.x < st) red[threadIdx.x] += red[threadIdx.x + st];
    __syncthreads();
  }
  if (threadIdx.x == 0) invn[blockIdx.x] = 1.0f / fmaxf(sqrtf(red[0]), 1e-12f);
}

// ---------------------------------------------------------------------------
// 5) S = (Q_raw K_rawT) scaled by invq[m]*invk[n].  M=N=256, K=2048 per head.
//    Block tile 64x32, 8 waves as 4x2.  Double-buffered async LDS staging:
//    per K-chunk: A 64xKC (8 KB), B 32xKC (4 KB); 768 b128 lane-slots,
//    3 async b128 instructions per thread per chunk.
// ---------------------------------------------------------------------------
__global__ void qk_kernel(const float* __restrict__ q, const float* __restrict__ kmat,
                          const float* __restrict__ invq, const float* __restrict__ invk,
                          float* __restrict__ S) {
  __shared__ float lA[2][64 * KC];
  __shared__ float lB[2][32 * KC];
  int tid  = threadIdx.x;
  int lane = tid & 31, wvid = tid >> 5;
  int mt = wvid >> 1, nt = wvid & 1;
  int z  = blockIdx.z;                       // b*HEADS + head
  int mb = blockIdx.y * 64;
  int nb = blockIdx.x * 32;
  int lm = lane & 15, kb = (lane >> 4) << 1;

  const float* qbase = q    + ((long)z * WW + mb) * DFEAT;
  const float* kbase = kmat + ((long)z * WW + nb) * DFEAT;

  v8f acc = {};
  const int nchunk = DFEAT / KC;             // 64

  // prologue: stage chunk 0 into buffer 0
  for (int i = 0; i < 3; ++i) {
    int s = tid + (i << 8);
    if (s < 512) {
      int row = s >> 3, seg = s & 7;
      async_ld16(qbase + (long)row * DFEAT + seg * 4, &lA[0][s * 4]);
    } else {
      int s2 = s - 512, row = s2 >> 3, seg = s2 & 7;
      async_ld16(kbase + (long)row * DFEAT + seg * 4, &lB[0][s2 * 4]);
    }
  }
  wait_async0();
  __syncthreads();

  for (int c = 0; c < nchunk; ++c) {
    int buf = c & 1;
    if (c + 1 < nchunk) {                    // stage next chunk into buf^1
      int k0 = (c + 1) * KC;
      for (int i = 0; i < 3; ++i) {
        int s = tid + (i << 8);
        if (s < 512) {
          int row = s >> 3, seg = s & 7;
          async_ld16(qbase + (long)row * DFEAT + k0 + seg * 4, &lA[buf ^ 1][s * 4]);
        } else {
          int s2 = s - 512, row = s2 >> 3, seg = s2 & 7;
          async_ld16(kbase + (long)row * DFEAT + k0 + seg * 4, &lB[buf ^ 1][s2 * 4]);
        }
      }
    }
    const float* Ar = &lA[buf][(mt * 16 + lm) * KC];
    const float* Br = &lB[buf][(nt * 16 + lm) * KC];
    for (int kk = 0; kk < KC; kk += 4) {
      v2f a = *(const v2f*)(Ar + kk + kb);
      v2f b = *(const v2f*)(Br + kk + kb);
      acc = wmma4(a, b, acc);
    }
    wait_async0();                           // own staging done ...
    __syncthreads();                         // ... and everyone done reading buf
  }

  int ng = nb + nt * 16 + lm;
  float sk = invk[(long)z * WW + ng];
  for (int r = 0; r < 8; ++r) {
    int mg = mb + mt * 16 + r + ((lane >> 4) << 3);
    S[((long)z * WW + mg) * WW + ng] = acc[r] * invq[(long)z * WW + mg] * sk;
  }
}

// ---------------------------------------------------------------------------
// 6) Row softmax of S (rows of length 256, one block per row).
// ---------------------------------------------------------------------------
__global__ void softmax_kernel(float* __restrict__ S) {
  __shared__ float red[256];
  float* r = S + (long)blockIdx.x * WW;
  int t = threadIdx.x;
  float v = r[t];
  red[t] = v; __syncthreads();
  for (int st = 128; st; st >>= 1) {
    if (t < st) red[t] = fmaxf(red[t], red[t + st]);
    __syncthreads();
  }
  float mx = red[0]; __syncthreads();
  float e = __expf(v - mx);
  red[t] = e; __syncthreads();
  for (int st = 128; st; st >>= 1) {
    if (t < st) red[t] += red[t + st];
    __syncthreads();
  }
  r[t] = e / red[0];
}

// ---------------------------------------------------------------------------
// 7) out4 = attn @ V + Q_normalized.  M=256(w), N=2048(feat), K=256 per head.
//    Block tile 32x64, 8 waves as 2x4.  Async LDS staging, double-buffered:
//    A 32xKC (4 KB) + B KCx64 (8 KB) per chunk; 3 b128 slots per thread.
// ---------------------------------------------------------------------------
__global__ void av_kernel(const float* __restrict__ S, const float* __restrict__ vmat,
                          const float* __restrict__ qraw, const float* __restrict__ invq,
                          float* __restrict__ o4) {
  __shared__ float lA[2][32 * KC];           // [m][k]
  __shared__ float lB[2][KC * 64];           // [k][n]
  int tid  = threadIdx.x;
  int lane = tid & 31, wvid = tid >> 5;
  int mt = wvid >> 2, nt = wvid & 3;
  int z  = blockIdx.z;
  int mb = blockIdx.y * 32;
  int nb = blockIdx.x * 64;
  int lm = lane & 15, kb = (lane >> 4) << 1;

  const float* abase = S + ((long)z * WW + mb) * WW;              // attn rows
  const float* vbase = vmat + (long)z * WW * DFEAT + nb;          // V cols slice

  v8f acc = {};
  const int nchunk = WW / KC;                // 8

  for (int i = 0; i < 3; ++i) {
    int s = tid + (i << 8);
    if (s < 256) {
      int row = s >> 3, seg = s & 7;
      async_ld16(abase + (long)row * WW + seg * 4, &lA[0][s * 4]);
    } else {
      int s2 = s - 256, row = s2 >> 4, seg = s2 & 15;
      async_ld16(vbase + (long)row * DFEAT + seg * 4, &lB[0][s2 * 4]);
    }
  }
  wait_async0();
  __syncthreads();

  for (int c = 0; c < nchunk; ++c) {
    int buf = c & 1;
    if (c + 1 < nchunk) {
      int k0 = (c + 1) * KC;
      for (int i = 0; i < 3; ++i) {
        int s = tid + (i << 8);
        if (s < 256) {
          int row = s >> 3, seg = s & 7;
          async_ld16(abase + (long)row * WW + k0 + seg * 4, &lA[buf ^ 1][s * 4]);
        } else {
          int s2 = s - 256, row = s2 >> 4, seg = s2 & 15;
          async_ld16(vbase + (long)(k0 + row) * DFEAT + seg * 4, &lB[buf ^ 1][s2 * 4]);
        }
      }
    }
    const float* Ar = &lA[buf][(mt * 16 + lm) * KC];
    const float* Bc = &lB[buf][nt * 16 + lm];
    for (int kk = 0; kk < KC; kk += 4) {
      v2f a = *(const v2f*)(Ar + kk + kb);
      v2f b;
      b.x = Bc[(kk + kb) * 64];
      b.y = Bc[(kk + kb + 1) * 64];
      acc = wmma4(a, b, acc);
    }
    wait_async0();
    __syncthreads();
  }

  int ng = nb + nt * 16 + lm;
  for (int r = 0; r < 8; ++r) {
    int mg = mb + mt * 16 + r + ((lane >> 4) << 3);
    long qi = ((long)z * WW + mg) * DFEAT + ng;
    o4[qi] = acc[r] + qraw[qi] * invq[(long)z * WW + mg];
  }
}

// ---------------------------------------------------------------------------
// 8) Final pointwise conv (gathers out4 from attention layout) + residual.
// ---------------------------------------------------------------------------
__global__ void final_kernel(const float* __restrict__ o4, const float* __restrict__ wpw,
                             const float* __restrict__ bpw, const float* __restrict__ x1n,
                             const float* __restrict__ x2n, float* __restrict__ out) {
  int lane = threadIdx.x & 31, wvid = threadIdx.x >> 5;
  int mtile = wvid & 3, nhalf = wvid >> 2;
  int w0 = blockIdx.x * 32 + nhalf * 16;
  int h  = blockIdx.y;
  int b  = blockIdx.z;
  int m0 = mtile * 16;
  int lm = lane & 15, kb = (lane >> 4) << 1;
  int wn = w0 + lm;
  v8f acc = {};
  for (int k0 = 0; k0 < DIM; k0 += 4) {
    v2f a, bb;
    a.x = wpw[(m0 + lm) * DIM + k0 + kb];
    a.y = wpw[(m0 + lm) * DIM + k0 + kb + 1];
    int c0 = k0 + kb, c1 = k0 + kb + 1;
    bb.x = o4[((long)(b * HEADS + (c0 >> 3)) * WW + wn) * DFEAT + h * CPH + (c0 & 7)];
    bb.y = o4[((long)(b * HEADS + (c1 >> 3)) * WW + wn) * DFEAT + h * CPH + (c1 & 7)];
    acc = wmma4(a, bb, acc);
  }
  for (int r = 0; r < 8; ++r) {
    int o = m0 + r + ((lane >> 4) << 3);
    long i = ((long)(b * DIM + o) * HH + h) * WW + wn;
    out[i] = acc[r] + bpw[o] + x1n[i] + x2n[i];
  }
}

// ---------------------------------------------------------------------------
extern "C" void kernel_launch(void* const* d_in, const int* in_sizes, int n_in,
                              void* d_out, int out_size, void* d_ws, size_t ws_size,
                              hipStream_t stream) {
  const float* x1   = (const float*)d_in[0];
  const float* x2   = (const float*)d_in[1];
  const float* ln1w = (const float*)d_in[2];
  const float* ln1b = (const float*)d_in[3];
  const float* ln2w = (const float*)d_in[4];
  const float* ln2b = (const float*)d_in[5];
  const float* bdw  = (const float*)d_in[6];   // (12, 64)
  const float* wpw  = (const float*)d_in[7];   // (64, 64)
  const float* bpw  = (const float*)d_in[8];   // (64,)
  const float* k11 = (const float*)d_in[9];
  const float* k12 = (const float*)d_in[10];
  const float* k13 = (const float*)d_in[11];
  const float* k14 = (const float*)d_in[12];
  const float* k15 = (const float*)d_in[13];
  const float* k16 = (const float*)d_in[14];
  const float* k21 = (const float*)d_in[15];
  const float* k22 = (const float*)d_in[16];
  const float* k23 = (const float*)d_in[17];
  const float* k24 = (const float*)d_in[18];
  const float* k25 = (const float*)d_in[19];
  const float* k26 = (const float*)d_in[20];
  float* out = (float*)d_out;

  float* W    = (float*)d_ws;
  float* x1n  = W;
  float* x2n  = W + 1 * NTOT;
  float* dt1  = W + 2 * NTOT;
  float* dt2  = W + 3 * NTOT;
  float* p1   = W + 4 * NTOT;   // attention layout (b,head,w,h*cph)
  float* p2   = W + 5 * NTOT;
  float* o4   = W + 6 * NTOT;
  float* S    = W + 7 * NTOT;   // (32, 256, 256)
  float* invq = S + (long)BATCH * HEADS * WW * WW;
  float* invk = invq + BATCH * HEADS * WW;

  ln_kernel<<<(BATCH * HW) / 256, 256, 0, stream>>>(x1, x2, ln1w, ln1b, ln2w, ln2b,
                                                    x1n, x2n);
  dim3 gdw(HH, BATCH * DIM);
  dw_kernel<<<gdw, 256, 0, stream>>>(x1n, dt1, k11, k12, k13, k14, k15, k16, bdw);
  dw_kernel<<<gdw, 256, 0, stream>>>(x2n, dt2, k21, k22, k23, k24, k25, k26,
                                     bdw + 6 * DIM);
  dim3 gpw(WW / 32, HH, BATCH);
  pw_attn_kernel<<<gpw, 256, 0, stream>>>(dt1, wpw, bpw, p1);
  pw_attn_kernel<<<gpw, 256, 0, stream>>>(dt2, wpw, bpw, p2);
  rownorm_kernel<<<BATCH * HEADS * WW, 256, 0, stream>>>(p1, invq);
  rownorm_kernel<<<BATCH * HEADS * WW, 256, 0, stream>>>(p2, invk);
  qk_kernel<<<dim3(WW / 32, WW / 64, BATCH * HEADS), 256, 0, stream>>>(p1, p2, invq,
                                                                       invk, S);
  softmax_kernel<<<BATCH * HEADS * WW, 256, 0, stream>>>(S);
  av_kernel<<<dim3(DFEAT / 64, WW / 32, BATCH * HEADS), 256, 0, stream>>>(S, p2, p1,
                                                                          invq, o4);
  final_kernel<<<gpw, 256, 0, stream>>>(o4, wpw, bpw, x1n, x2n, out);
}